// dilatedweightBCE_10505490006429
// MI455X (gfx1250) — compile-verified
//
#include <hip/hip_runtime.h>

#define IMG_H 1024
#define IMG_W 1024
#define BATCH 16
#define ROWS_PER_BLK 8
#define TPB 256
#define STRIPS_PER_IMG (IMG_H / ROWS_PER_BLK)            // 128
#define NBLK (BATCH * STRIPS_PER_IMG)                    // 2048
#define N_TOTAL (BATCH * IMG_H * IMG_W)                  // 16777216 = 2^24

typedef __attribute__((ext_vector_type(2))) float v2f;
typedef __attribute__((ext_vector_type(8))) float v8f;

__device__ __forceinline__ float bce_term(float x, float t, float dil) {
    // weight: edge -> 20, dilated band -> 5, else 1
    float w = (t > 0.5f) ? 20.0f : ((dil > 0.5f) ? 5.0f : 1.0f);
    float ax = __builtin_fabsf(x);
    // log1p(exp(-|x|)): arg of log is in [1,2] -> plain log is exact enough
    float sp = __logf(1.0f + __expf(-ax));
    return w * (fmaxf(x, 0.0f) - x * t + sp);
}

__global__ __launch_bounds__(TPB) void bce_main_kernel(const float* __restrict__ pred,
                                                       const float* __restrict__ tgt,
                                                       float* __restrict__ partials) {
    const int bid   = blockIdx.x;
    const int tid   = threadIdx.x;
    const int b     = bid / STRIPS_PER_IMG;
    const int strip = bid % STRIPS_PER_IMG;
    const int r0    = strip * ROWS_PER_BLK;
    const int c0    = tid * 4;                       // 0..1020, 16B aligned

    const float* __restrict__ timg = tgt  + (size_t)b * IMG_H * IMG_W;
    const float* __restrict__ pimg = pred + (size_t)b * IMG_H * IMG_W;

    // Rotating 3-row stencil window: up / cur / down, each = float4 + L/R edge scalars.
    float4 tu, tc, td;
    float  lu, lc, ld, ru, rc, rd;

    auto load4 = [&](int r) -> float4 {
        return *reinterpret_cast<const float4*>(timg + (size_t)r * IMG_W + c0);
    };
    auto loadL = [&](int r) -> float {
        return (c0 > 0) ? timg[(size_t)r * IMG_W + (c0 - 1)] : 0.0f;
    };
    auto loadR = [&](int r) -> float {
        return (c0 + 4 < IMG_W) ? timg[(size_t)r * IMG_W + (c0 + 4)] : 0.0f;
    };

    if (r0 > 0) { tu = load4(r0 - 1); lu = loadL(r0 - 1); ru = loadR(r0 - 1); }
    else        { tu = make_float4(0.f, 0.f, 0.f, 0.f); lu = 0.f; ru = 0.f; }
    tc = load4(r0); lc = loadL(r0); rc = loadR(r0);

    float acc = 0.0f;

#pragma unroll
    for (int ri = 0; ri < ROWS_PER_BLK; ++ri) {
        const int r = r0 + ri;
        if (r + 1 < IMG_H) { td = load4(r + 1); ld = loadL(r + 1); rd = loadR(r + 1); }
        else               { td = make_float4(0.f, 0.f, 0.f, 0.f); ld = 0.f; rd = 0.f; }

        const float4 p = *reinterpret_cast<const float4*>(pimg + (size_t)r * IMG_W + c0);

        // vertical max of the 3 rows (OOB/-inf padding == 0 for a {0,1} target)
        const float vm0 = fmaxf(fmaxf(tu.x, tc.x), td.x);
        const float vm1 = fmaxf(fmaxf(tu.y, tc.y), td.y);
        const float vm2 = fmaxf(fmaxf(tu.z, tc.z), td.z);
        const float vm3 = fmaxf(fmaxf(tu.w, tc.w), td.w);
        const float vml = fmaxf(fmaxf(lu, lc), ld);
        const float vmr = fmaxf(fmaxf(ru, rc), rd);

        // horizontal max of 3 -> dilated
        const float d0 = fmaxf(fmaxf(vml, vm0), vm1);
        const float d1 = fmaxf(fmaxf(vm0, vm1), vm2);
        const float d2 = fmaxf(fmaxf(vm1, vm2), vm3);
        const float d3 = fmaxf(fmaxf(vm2, vm3), vmr);

        acc += bce_term(p.x, tc.x, d0);
        acc += bce_term(p.y, tc.y, d1);
        acc += bce_term(p.z, tc.z, d2);
        acc += bce_term(p.w, tc.w, d3);

        // rotate window
        tu = tc; lu = lc; ru = rc;
        tc = td; lc = ld; rc = rd;
    }

    // ---- wave32 reduction via V_WMMA_F32_16X16X4_F32 (exact f32 FMA path) ----
    // A (16x4) = all ones: both A VGPRs = 1.0.
    // B (4x16): VGPR0 = per-lane partial (K=0 rows for lanes 0-15, K=2 for lanes 16-31),
    //           VGPR1 = 0  ->  D[m][n] = partial[n] + partial[n+16] for every m.
    v2f av; av[0] = 1.0f; av[1] = 1.0f;
    v2f bv; bv[0] = acc;  bv[1] = 0.0f;
    v8f cz = {0.f, 0.f, 0.f, 0.f, 0.f, 0.f, 0.f, 0.f};
    v8f dm = __builtin_amdgcn_wmma_f32_16x16x4_f32(
        /*neg_a=*/false, av, /*neg_b=*/false, bv,
        /*c_mod=*/(short)0, cz, /*reuse_a=*/false, /*reuse_b=*/false);

    float wsum = dm[0];                 // lanes 0-15: D[0][n]; lanes 16-31: D[8][n] (identical)
    wsum += __shfl_xor(wsum, 1, 32);    // sum the 16 columns within each half
    wsum += __shfl_xor(wsum, 2, 32);
    wsum += __shfl_xor(wsum, 4, 32);
    wsum += __shfl_xor(wsum, 8, 32);    // every lane now holds the full wave sum

    __shared__ float smem[TPB / 32];
    const int wave = tid >> 5;
    const int lane = tid & 31;
    if (lane == 0) smem[wave] = wsum;
    __syncthreads();
    if (tid == 0) {
        float s = 0.0f;
#pragma unroll
        for (int i = 0; i < TPB / 32; ++i) s += smem[i];   // fixed order -> deterministic
        partials[bid] = s;
    }
}

__global__ __launch_bounds__(TPB) void bce_final_kernel(const float* __restrict__ partials,
                                                        float* __restrict__ out) {
    __shared__ float smem[TPB];
    const int tid = threadIdx.x;
    float s = 0.0f;
#pragma unroll
    for (int i = tid; i < NBLK; i += TPB) s += partials[i];  // fixed order per thread
    smem[tid] = s;
    __syncthreads();
#pragma unroll
    for (int off = TPB / 2; off > 0; off >>= 1) {            // fixed-shape tree
        if (tid < off) smem[tid] += smem[tid + off];
        __syncthreads();
    }
    if (tid == 0) out[0] = smem[0] * (1.0f / (float)N_TOTAL);  // *2^-24, exact scale
}

extern "C" void kernel_launch(void* const* d_in, const int* in_sizes, int n_in,
                              void* d_out, int out_size, void* d_ws, size_t ws_size,
                              hipStream_t stream) {
    const float* pred = (const float*)d_in[0];
    const float* tgt  = (const float*)d_in[1];
    float* partials   = (float*)d_ws;          // NBLK floats = 8 KiB scratch
    float* out        = (float*)d_out;

    bce_main_kernel<<<NBLK, TPB, 0, stream>>>(pred, tgt, partials);
    bce_final_kernel<<<1, TPB, 0, stream>>>(partials, out);
}